// Net_89429809037844
// MI455X (gfx1250) — compile-verified
//
#include <hip/hip_runtime.h>
#include <math.h>

typedef float v2f __attribute__((ext_vector_type(2)));
typedef float v8f __attribute__((ext_vector_type(8)));
typedef int   v4i __attribute__((ext_vector_type(4)));

#define HDIM   128
#define NB     64            // graphs
#define NPG0   2048
#define NEDGE  524288        // 64 * 2048 * 4
#define NN0    131072        // 64 * 2048
#define KK1    1639
#define KK2    1312
#define KK3    1050
#define NACT   124

static inline int cdiv(long a, long b) { return (int)((a + b - 1) / b); }

// ---------------------------------------------------------------------------
// WMMA fp32 GEMM:  C[M,128] = A[M,K] @ W[K,128]   (K multiple of 4, M mult 16)
// Block = 256 threads = 8 waves; each wave owns one 16-wide N tile.
// The 16xK A tile is staged once per block into LDS with the gfx1250 async
// global->LDS copy (ASYNCcnt) so the 8 waves don't each re-fetch it through
// the VMEM return path.
// ---------------------------------------------------------------------------
__global__ __launch_bounds__(256)
void gemm_wmma_n128(const float* __restrict__ A, const float* __restrict__ W,
                    float* __restrict__ C, int M, int K) {
  __shared__ float Atile[16 * HDIM];           // up to 8 KB
  const int tid  = threadIdx.x;
  const int tile = blockIdx.x;                 // M/16 tiles
  const int lane = tid & 31;
  const int wv   = tid >> 5;                   // 0..7 -> N tile
  const int m15  = lane & 15;
  const int hi   = lane >> 4;                  // 0/1
  const int kk   = hi << 1;                    // 0 or 2
  const int ncol = (wv << 4) + m15;

  const int nelem = 16 * K;                    // 64 (K=4) or 2048 (K=128)
  const float* Ablk = A + (size_t)(tile << 4) * K;

#if __has_builtin(__builtin_amdgcn_global_load_async_to_lds_b128) && \
    __has_builtin(__builtin_amdgcn_s_wait_asynccnt)
  for (int off = tid * 4; off < nelem; off += 256 * 4) {
    __builtin_amdgcn_global_load_async_to_lds_b128(
        (__attribute__((address_space(1))) v4i*)(Ablk + off),
        (__attribute__((address_space(3))) v4i*)(Atile + off),
        /*offset=*/0, /*cpol=*/0);
  }
  __builtin_amdgcn_s_wait_asynccnt(0);
  __syncthreads();
#else
  for (int off = tid * 4; off < nelem; off += 256 * 4) {
    *(float4*)(Atile + off) = *(const float4*)(Ablk + off);
  }
  __syncthreads();
#endif

  const float* arow = Atile + m15 * K;         // per-lane A row (LDS)
  v8f acc = {};
  for (int kb = 0; kb < K; kb += 4) {
    v2f a, b;
    a.x = arow[kb + kk];
    a.y = arow[kb + kk + 1];
    b.x = W[(size_t)(kb + kk)     * HDIM + ncol];
    b.y = W[(size_t)(kb + kk + 1) * HDIM + ncol];
    acc = __builtin_amdgcn_wmma_f32_16x16x4_f32(
        /*neg_a=*/false, a, /*neg_b=*/false, b,
        /*c_mod=*/(short)0, acc, /*reuse_a=*/false, /*reuse_b=*/false);
  }
  // D layout: VGPR v, lanes0-15 -> M=v ; lanes16-31 -> M=v+8
  float* cp = C + (size_t)((tile << 4) + (hi << 3)) * HDIM + ncol;
#pragma unroll
  for (int v = 0; v < 8; ++v) cp[(size_t)v * HDIM] = acc[v];
}

// ---------------------------------------------------------------------------
// small utility kernels
// ---------------------------------------------------------------------------
__global__ void init_f32_kernel(float* __restrict__ p, int n, float v) {
  int i = blockIdx.x * blockDim.x + threadIdx.x;
  if (i < n) p[i] = v;
}
__global__ void init_i32_kernel(int* __restrict__ p, int n, int v) {
  int i = blockIdx.x * blockDim.x + threadIdx.x;
  if (i < n) p[i] = v;
}
__global__ void rsqrt_kernel(const float* __restrict__ deg, float* __restrict__ dis, int n) {
  int i = blockIdx.x * blockDim.x + threadIdx.x;
  if (i < n) dis[i] = rsqrtf(deg[i]);
}
__global__ void relu_kernel(float* __restrict__ x, int n) {
  int i = blockIdx.x * blockDim.x + threadIdx.x;
  if (i < n) x[i] = fmaxf(x[i], 0.f);
}

// deg[dst] += valid (valid==nullptr -> all valid); deg pre-initialized to 1.0 (self loop)
__global__ void deg_scatter_kernel(const int* __restrict__ dst, const int* __restrict__ valid,
                                   float* __restrict__ deg, int nE) {
  int e = blockIdx.x * blockDim.x + threadIdx.x;
  if (e >= nE) return;
  if (valid && !valid[e]) return;
  atomicAdd(&deg[dst[e]], 1.0f);
}

// out[i,f] = h[i,f]*dis[i]^2 + b[f]   (self-loop term + bias)
__global__ void init_out_kernel(const float* __restrict__ h, const float* __restrict__ dis,
                                const float* __restrict__ b, float* __restrict__ out, int n) {
  int q = blockIdx.x * blockDim.x + threadIdx.x;
  if (q >= n * HDIM) return;
  int i = q >> 7, f = q & (HDIM - 1);
  float ds = dis[i];
  out[q] = h[q] * ds * ds + b[f];
}

// out[dst] += h[src] * dis[src]*dis[dst]  (4 feats per thread, f32 atomics, L2-resident)
__global__ void agg_scatter_kernel(const float* __restrict__ h, const int* __restrict__ s,
                                   const int* __restrict__ d, const int* __restrict__ valid,
                                   const float* __restrict__ dis, float* __restrict__ out, int nE) {
  int q = blockIdx.x * blockDim.x + threadIdx.x;
  int e = q >> 5;                              // HDIM/4 = 32 groups per edge
  if (e >= nE) return;
  if (valid && !valid[e]) return;
  int fg = (q & 31) << 2;
  int si = s[e], di = d[e];
  float c = dis[si] * dis[di];
  const float4 hv = *(const float4*)(h + (size_t)si * HDIM + fg);
  float* op = out + (size_t)di * HDIM + fg;
  atomicAdd(op + 0, hv.x * c);
  atomicAdd(op + 1, hv.y * c);
  atomicAdd(op + 2, hv.z * c);
  atomicAdd(op + 3, hv.w * c);
}

// ---------------------------------------------------------------------------
// TopK pooling: one block per graph. Bitonic sort of 2048 64-bit keys in LDS.
// key = (~orderbits(score) << 32) | idx  -> ascending sort = desc score, asc idx.
// ---------------------------------------------------------------------------
__global__ __launch_bounds__(1024)
void topk_pool_kernel(const float* __restrict__ h, const float* __restrict__ p,
                      float* __restrict__ xnew, int* __restrict__ new_pos,
                      int npg, int k) {
  __shared__ unsigned long long keys[2048];
  const int g = blockIdx.x;
  const int t = threadIdx.x;

  float s2 = 0.f;
  for (int f = 0; f < HDIM; ++f) { float pv = p[f]; s2 += pv * pv; }
  const float invn = rsqrtf(s2);

  const float* hg = h + (size_t)g * npg * HDIM;
  for (int i = t; i < 2048; i += 1024) {
    float sc = -2.0f;                          // below tanh range -> sorts last
    if (i < npg) {
      float dt = 0.f;
      const float* hr = hg + (size_t)i * HDIM;
      for (int f = 0; f < HDIM; ++f) dt += hr[f] * p[f];
      sc = tanhf(dt * invn);
    }
    unsigned u = __float_as_uint(sc);
    u = (u & 0x80000000u) ? ~u : (u | 0x80000000u);      // monotone ascending bits
    keys[i] = ((unsigned long long)(~u) << 32) | (unsigned)i;
  }
  __syncthreads();

  for (int size = 2; size <= 2048; size <<= 1) {
    for (int stride = size >> 1; stride > 0; stride >>= 1) {
      for (int ii = t; ii < 2048; ii += 1024) {
        int jj = ii ^ stride;
        if (jj > ii) {
          bool asc = ((ii & size) == 0);
          unsigned long long a = keys[ii], b = keys[jj];
          if (asc ? (a > b) : (a < b)) { keys[ii] = b; keys[jj] = a; }
        }
      }
      __syncthreads();
    }
  }

  for (int j = t; j < k; j += 1024) {
    unsigned idx = (unsigned)(keys[j] & 0xffffffffu);
    new_pos[(size_t)g * npg + idx] = g * k + j;
  }
  const int total = k * HDIM;
  for (int q = t; q < total; q += 1024) {
    int j = q >> 7, f = q & (HDIM - 1);
    unsigned long long kk = keys[j];
    unsigned idx = (unsigned)(kk & 0xffffffffu);
    unsigned u = ~(unsigned)(kk >> 32);
    unsigned raw = (u & 0x80000000u) ? (u & 0x7fffffffu) : ~u;   // un-order
    float val = __uint_as_float(raw);
    xnew[(size_t)(g * k + j) * HDIM + f] = hg[(size_t)idx * HDIM + f] * val;
  }
}

// remap edges through new_pos; v_in nullable (all valid)
__global__ void remap_edges_kernel(const int* __restrict__ si, const int* __restrict__ di,
                                   const int* __restrict__ vi, const int* __restrict__ new_pos,
                                   int* __restrict__ so, int* __restrict__ dd,
                                   int* __restrict__ vo, int nE) {
  int e = blockIdx.x * blockDim.x + threadIdx.x;
  if (e >= nE) return;
  int v = vi ? vi[e] : 1;
  int ns = new_pos[si[e]];
  int nd = new_pos[di[e]];
  v = (v && ns >= 0 && nd >= 0) ? 1 : 0;
  so[e] = v ? ns : 0;
  dd[e] = v ? nd : 0;
  vo[e] = v;
}

// readout: zsum[g, 0:128] += max over k rows ; zsum[g, 128:256] += mean
__global__ void readout_kernel(const float* __restrict__ x, float* __restrict__ zsum, int k) {
  int g = blockIdx.x, f = threadIdx.x;         // 128 threads
  float mx = -3.4e38f, sm = 0.f;
  const float* xg = x + (size_t)g * k * HDIM + f;
  for (int j = 0; j < k; ++j) {
    float v = xg[(size_t)j * HDIM];
    mx = fmaxf(mx, v);
    sm += v;
  }
  zsum[g * 256 + f]       += mx;
  zsum[g * 256 + 128 + f] += sm / (float)k;
}

// dense MLP layer: out[64,fo] = act(in[64,fi] @ w[fi,fo] + b)  (act 0=relu, 1=sigmoid)
__global__ void mlp_kernel(const float* __restrict__ in, const float* __restrict__ w,
                           const float* __restrict__ bias, float* __restrict__ out,
                           int fi, int fo, int act) {
  int q = blockIdx.x * blockDim.x + threadIdx.x;
  if (q >= NB * fo) return;
  int g = q / fo, o = q % fo;
  float s = bias[o];
  const float* ig = in + (size_t)g * fi;
  for (int i = 0; i < fi; ++i) s += ig[i] * w[(size_t)i * fo + o];
  out[q] = (act == 0) ? fmaxf(s, 0.f) : 1.f / (1.f + expf(-s));
}

// ---------------------------------------------------------------------------
extern "C" void kernel_launch(void* const* d_in, const int* in_sizes, int n_in,
                              void* d_out, int out_size, void* d_ws, size_t ws_size,
                              hipStream_t stream) {
  const float* x   = (const float*)d_in[0];
  const int*   src = (const int*)  d_in[1];
  const int*   dst = (const int*)  d_in[2];
  const float* W1  = (const float*)d_in[3];
  const float* b1  = (const float*)d_in[4];
  const float* W2  = (const float*)d_in[5];
  const float* b2  = (const float*)d_in[6];
  const float* W3  = (const float*)d_in[7];
  const float* b3  = (const float*)d_in[8];
  const float* p1  = (const float*)d_in[9];
  const float* p2  = (const float*)d_in[10];
  const float* p3  = (const float*)d_in[11];
  const float* lw1 = (const float*)d_in[12];
  const float* lb1 = (const float*)d_in[13];
  const float* lw2 = (const float*)d_in[14];
  const float* lb2 = (const float*)d_in[15];
  const float* lw3 = (const float*)d_in[16];
  const float* lb3 = (const float*)d_in[17];
  float* out = (float*)d_out;

  // workspace layout (all f32/i32); ~206 MB total, fully L2-resident working set
  const size_t FB = (size_t)NN0 * HDIM;        // 16.7M floats per feature buffer
  float* F0  = (float*)d_ws;
  float* F1  = F0 + FB;
  float* F2  = F1 + FB;
  float* DEG = F2 + FB;
  float* DIS = DEG + NN0;
  int*   NP  = (int*)(DIS + NN0);
  int*   sA  = NP + NN0;
  int*   dA  = sA + NEDGE;
  int*   vA  = dA + NEDGE;
  int*   sB  = vA + NEDGE;
  int*   dB  = sB + NEDGE;
  int*   vB  = dB + NEDGE;
  float* ZS  = (float*)(vB + NEDGE);           // [64,256]
  float* Z1  = ZS + NB * 256;                  // [64,128]
  float* Z2  = Z1 + NB * HDIM;                 // [64,64]

  const int T = 256;
  const int n1 = NN0;                          // 131072
  const int n2 = NB * KK1;                     // 104896
  const int n3 = NB * KK2;                     // 83968

  init_f32_kernel<<<cdiv(NB * 256, T), T, 0, stream>>>(ZS, NB * 256, 0.f);

  // ================= Level 1 (K=4 input features) =================
  gemm_wmma_n128<<<n1 / 16, T, 0, stream>>>(x, W1, F0, n1, 4);
  init_f32_kernel<<<cdiv(n1, T), T, 0, stream>>>(DEG, n1, 1.0f);
  deg_scatter_kernel<<<cdiv(NEDGE, T), T, 0, stream>>>(dst, nullptr, DEG, NEDGE);
  rsqrt_kernel<<<cdiv(n1, T), T, 0, stream>>>(DEG, DIS, n1);
  init_out_kernel<<<cdiv((long)n1 * HDIM, T), T, 0, stream>>>(F0, DIS, b1, F1, n1);
  agg_scatter_kernel<<<cdiv((long)NEDGE * 32, T), T, 0, stream>>>(F0, src, dst, nullptr, DIS, F1, NEDGE);
  relu_kernel<<<cdiv((long)n1 * HDIM, T), T, 0, stream>>>(F1, n1 * HDIM);
  init_i32_kernel<<<cdiv(n1, T), T, 0, stream>>>(NP, n1, -1);
  topk_pool_kernel<<<NB, 1024, 0, stream>>>(F1, p1, F2, NP, NPG0, KK1);
  remap_edges_kernel<<<cdiv(NEDGE, T), T, 0, stream>>>(src, dst, nullptr, NP, sA, dA, vA, NEDGE);
  readout_kernel<<<NB, HDIM, 0, stream>>>(F2, ZS, KK1);

  // ================= Level 2 (K=128) =================
  gemm_wmma_n128<<<n2 / 16, T, 0, stream>>>(F2, W2, F0, n2, HDIM);
  init_f32_kernel<<<cdiv(n2, T), T, 0, stream>>>(DEG, n2, 1.0f);
  deg_scatter_kernel<<<cdiv(NEDGE, T), T, 0, stream>>>(dA, vA, DEG, NEDGE);
  rsqrt_kernel<<<cdiv(n2, T), T, 0, stream>>>(DEG, DIS, n2);
  init_out_kernel<<<cdiv((long)n2 * HDIM, T), T, 0, stream>>>(F0, DIS, b2, F1, n2);
  agg_scatter_kernel<<<cdiv((long)NEDGE * 32, T), T, 0, stream>>>(F0, sA, dA, vA, DIS, F1, NEDGE);
  relu_kernel<<<cdiv((long)n2 * HDIM, T), T, 0, stream>>>(F1, n2 * HDIM);
  init_i32_kernel<<<cdiv(n2, T), T, 0, stream>>>(NP, n2, -1);
  topk_pool_kernel<<<NB, 1024, 0, stream>>>(F1, p2, F0, NP, KK1, KK2);   // xnew -> F0
  remap_edges_kernel<<<cdiv(NEDGE, T), T, 0, stream>>>(sA, dA, vA, NP, sB, dB, vB, NEDGE);
  readout_kernel<<<NB, HDIM, 0, stream>>>(F0, ZS, KK2);

  // ================= Level 3 (K=128) =================
  gemm_wmma_n128<<<n3 / 16, T, 0, stream>>>(F0, W3, F2, n3, HDIM);       // h -> F2
  init_f32_kernel<<<cdiv(n3, T), T, 0, stream>>>(DEG, n3, 1.0f);
  deg_scatter_kernel<<<cdiv(NEDGE, T), T, 0, stream>>>(dB, vB, DEG, NEDGE);
  rsqrt_kernel<<<cdiv(n3, T), T, 0, stream>>>(DEG, DIS, n3);
  init_out_kernel<<<cdiv((long)n3 * HDIM, T), T, 0, stream>>>(F2, DIS, b3, F1, n3);
  agg_scatter_kernel<<<cdiv((long)NEDGE * 32, T), T, 0, stream>>>(F2, sB, dB, vB, DIS, F1, NEDGE);
  relu_kernel<<<cdiv((long)n3 * HDIM, T), T, 0, stream>>>(F1, n3 * HDIM);
  init_i32_kernel<<<cdiv(n3, T), T, 0, stream>>>(NP, n3, -1);
  topk_pool_kernel<<<NB, 1024, 0, stream>>>(F1, p3, F2, NP, KK2, KK3);   // xnew -> F2
  readout_kernel<<<NB, HDIM, 0, stream>>>(F2, ZS, KK3);

  // ================= MLP head =================
  mlp_kernel<<<cdiv(NB * HDIM, T), T, 0, stream>>>(ZS, lw1, lb1, Z1, 256, HDIM, 0);
  mlp_kernel<<<cdiv(NB * 64, T), T, 0, stream>>>(Z1, lw2, lb2, Z2, HDIM, 64, 0);
  mlp_kernel<<<cdiv(NB * NACT, T), T, 0, stream>>>(Z2, lw3, lb3, out, 64, NACT, 1);
}